// HBNS_38723425140758
// MI455X (gfx1250) — compile-verified
//
#include <hip/hip_runtime.h>

// ---------------- problem constants ----------------
constexpr int N    = 50000;      // NS == NT
constexpr int E    = 1600000;
constexpr int D    = 64;
constexpr long N64 = (long)N * D;            // 3.2M floats per feature table
constexpr int KTOT = 576;                    // 3 parts (sum,mean,max) * 3 branches * 64

// workspace layout (floats):
//  [0..3]*N64   : s_msg, s_msg2, t_msg, t_msg2
//  [4..6]*N64   : sum_src (3 branches)
//  [7..9]*N64   : sum_tgt
//  [10..12]*N64 : max_src
//  [13..15]*N64 : max_tgt
//  16*N64       : cnt_src[N], cnt_tgt[N]
//  16*N64+2N    : Bp[2][576][64] rearranged weights

typedef __attribute__((ext_vector_type(2))) float v2f;
typedef __attribute__((ext_vector_type(8))) float v8f;

// CDNA5-native non-returning FP atomics (device scope, tracked on STOREcnt).
__device__ __forceinline__ void gadd(float* p, float v) {
    asm volatile("global_atomic_add_f32 %0, %1, off scope:SCOPE_DEV"
                 :: "v"(p), "v"(v) : "memory");
}
__device__ __forceinline__ void gmax(float* p, float v) {
    asm volatile("global_atomic_max_num_f32 %0, %1, off scope:SCOPE_DEV"
                 :: "v"(p), "v"(v) : "memory");
}

// ---------------- init accumulators ----------------
__global__ void k_init(float* __restrict__ ws) {
    long idx = (long)blockIdx.x * blockDim.x + threadIdx.x;
    if (idx < 6 * N64) {
        ws[4 * N64 + idx]  = 0.0f;                 // sums (src then tgt)
        ws[10 * N64 + idx] = -__builtin_inff();    // maxes
    }
    if (idx < 2 * (long)N) ws[16 * N64 + idx] = 0.0f;  // counts
}

// ---------------- rearrange [3,192,64] weights into [576,64] ----------------
__global__ void k_prep(const float* __restrict__ srcW, const float* __restrict__ tgtW,
                       float* __restrict__ Bp) {
    int idx = blockIdx.x * blockDim.x + threadIdx.x;
    if (idx >= 2 * KTOT * D) return;
    int side = idx / (KTOT * D);
    int rem  = idx - side * KTOT * D;
    int kk   = rem >> 6;          // 0..575
    int n    = rem & 63;
    int part = kk / 192;          // 0=sum row block, 1=mean, 2=max
    int k3   = kk - part * 192;
    int kmat = k3 >> 6;           // branch 0..2
    int kin  = k3 & 63;
    const float* W = side ? tgtW : srcW;   // [3, 192, 64]
    Bp[idx] = W[(long)kmat * 192 * 64 + (long)(part * 64 + kin) * 64 + n];
}

// ---------------- dense projections via fp32 WMMA ----------------
// one wave computes a 16x64 tile of Y = X[50000,64] @ W[64,64]
__global__ void __launch_bounds__(128) k_proj(const float* __restrict__ xs,
                                              const float* __restrict__ xt,
                                              const float* __restrict__ w_s,
                                              const float* __restrict__ w_t,
                                              const float* __restrict__ w_s_cci,
                                              const float* __restrict__ w_t_cci,
                                              float* __restrict__ ws) {
    const int wave = threadIdx.x >> 5;
    const int lane = threadIdx.x & 31;
    const int rt   = blockIdx.x * 4 + wave;     // row tile, 0..3124
    if (rt >= N / 16) return;                    // wave-uniform: EXEC stays all-ones
    const int mat  = blockIdx.y;                 // 0:s_msg 1:s_msg2 2:t_msg 3:t_msg2
    const float* X = (mat < 2) ? xs : xt;
    const float* W = (mat == 0) ? w_s : (mat == 1) ? w_s_cci : (mat == 2) ? w_t : w_t_cci;
    float* Y = ws + (long)mat * N64;

    const int row = lane & 15;
    const int hi  = lane >> 4;                   // K-pair select per ISA fp32 A/B layout
    v8f acc[4] = {};
    for (int kk = 0; kk < 16; ++kk) {
        int k0 = kk * 4 + 2 * hi;
        v2f a;
        a.x = X[(long)(rt * 16 + row) * D + k0];
        a.y = X[(long)(rt * 16 + row) * D + k0 + 1];
#pragma unroll
        for (int ct = 0; ct < 4; ++ct) {
            v2f b;
            b.x = W[(long)k0 * D + ct * 16 + row];
            b.y = W[(long)(k0 + 1) * D + ct * 16 + row];
            acc[ct] = __builtin_amdgcn_wmma_f32_16x16x4_f32(
                false, a, false, b, (short)0, acc[ct], false, false);
        }
    }
#pragma unroll
    for (int ct = 0; ct < 4; ++ct)
#pragma unroll
        for (int j = 0; j < 8; ++j)
            Y[(long)(rt * 16 + j + 8 * hi) * D + ct * 16 + row] = acc[ct][j];
}

// ---------------- edge scatter: one wave per edge, lane = 2 adjacent channels ----------------
__global__ void k_scatter(const int* __restrict__ rows, const int* __restrict__ cols,
                          const float* __restrict__ nbv, const float* __restrict__ cciv,
                          float* __restrict__ ws) {
    long gid = (long)blockIdx.x * blockDim.x + threadIdx.x;
    long e   = gid >> 5;
    int lane = (int)(gid & 31);
    if (e >= E) return;
    int r = rows[e], c = cols[e];
    float a = nbv[e], b = cciv[e], h = a * b;

    const float2* s1 = (const float2*)(ws)           + (long)c * (D / 2);  // s_msg row
    const float2* s2 = (const float2*)(ws + N64)     + (long)c * (D / 2);  // s_msg2 row
    const float2* t1 = (const float2*)(ws + 2 * N64) + (long)r * (D / 2);  // t_msg row
    const float2* t2 = (const float2*)(ws + 3 * N64) + (long)r * (D / 2);  // t_msg2 row
    float* sum_src = ws + 4 * N64;
    float* sum_tgt = ws + 7 * N64;
    float* max_src = ws + 10 * N64;
    float* max_tgt = ws + 13 * N64;
    float* cnt     = ws + 16 * N64;

    float2 tm  = t1[lane];
    float2 tm2 = t2[lane];
    float2 sm  = s1[lane];
    float2 sm2 = s2[lane];

    long oc  = (long)c * D + 2 * lane;     // src-side segment, channel pair base
    long orr = (long)r * D + 2 * lane;     // tgt-side segment

#pragma unroll
    for (int u = 0; u < 2; ++u) {
        float tmu  = u ? tm.y  : tm.x;
        float tm2u = u ? tm2.y : tm2.x;
        float smu  = u ? sm.y  : sm.x;
        float sm2u = u ? sm2.y : sm2.x;
        float g0 = tmu * a, g1 = tm2u * b, g2 = tm2u * h;
        gadd(&sum_src[oc + u], g0);
        gadd(&sum_src[N64 + oc + u], g1);
        gadd(&sum_src[2 * N64 + oc + u], g2);
        gmax(&max_src[oc + u], g0);
        gmax(&max_src[N64 + oc + u], g1);
        gmax(&max_src[2 * N64 + oc + u], g2);
        float p0 = smu * a, p1 = sm2u * b, p2 = sm2u * h;
        gadd(&sum_tgt[orr + u], p0);
        gadd(&sum_tgt[N64 + orr + u], p1);
        gadd(&sum_tgt[2 * N64 + orr + u], p2);
        gmax(&max_tgt[orr + u], p0);
        gmax(&max_tgt[N64 + orr + u], p1);
        gmax(&max_tgt[2 * N64 + orr + u], p2);
    }
    if (lane == 0) {
        gadd(&cnt[c], 1.0f);
        gadd(&cnt[N + r], 1.0f);
    }
}

// ---------------- finalize: [16 nodes, 576] @ Bp[576,64] via WMMA ----------------
__global__ void __launch_bounds__(128) k_final(const float* __restrict__ ws,
                                               const float* __restrict__ src_b,
                                               const float* __restrict__ tgt_b,
                                               float* __restrict__ out) {
    __shared__ float As[16][KTOT + 8];
    const int tid  = threadIdx.x;
    const int t    = blockIdx.x;        // node tile 0..3124
    const int side = blockIdx.y;        // 0 = msg_src, 1 = msg_tgt
    const float* sums = ws + (4 + 3 * side) * N64;
    const float* maxs = ws + (10 + 3 * side) * N64;
    const float* cnt  = ws + 16 * N64 + (long)side * N;
    const float* Bp   = ws + 16 * N64 + 2 * (long)N + (long)side * KTOT * D;
    const float* bias = side ? tgt_b : src_b;
    out += (long)side * N64;

    // stage 1: build A rows = [sum(192) | mean(192) | max'(192)]
    for (int idx = tid; idx < 16 * KTOT; idx += 128) {
        int row  = idx / KTOT;
        int kk   = idx - row * KTOT;
        int node = t * 16 + row;
        int part = kk / 192;
        int k3   = kk - part * 192;
        int kmat = k3 >> 6;
        int kin  = k3 & 63;
        long off = (long)kmat * N64 + (long)node * D + kin;
        float v;
        if (part == 0)      v = sums[off];
        else if (part == 1) v = sums[off] / fmaxf(cnt[node], 1.0f);
        else { float m = maxs[off]; v = (m == -__builtin_inff()) ? 0.0f : m; }
        As[row][kk] = v;
    }
    __syncthreads();

    // stage 2: 4 waves, one 16-col output tile each; K = 576 in 144 WMMA steps
    const int wave = tid >> 5;
    const int lane = tid & 31;
    const int row  = lane & 15;
    const int hi   = lane >> 4;
    const int n    = wave * 16 + row;
    v8f acc = {};
    for (int kk = 0; kk < KTOT / 4; ++kk) {
        int k0 = kk * 4 + 2 * hi;
        v2f a; a.x = As[row][k0];            a.y = As[row][k0 + 1];
        v2f b; b.x = Bp[(long)k0 * D + n];   b.y = Bp[(long)(k0 + 1) * D + n];
        acc = __builtin_amdgcn_wmma_f32_16x16x4_f32(
            false, a, false, b, (short)0, acc, false, false);
    }
    float bs = bias[n] + bias[64 + n] + bias[128 + n];
#pragma unroll
    for (int j = 0; j < 8; ++j)
        out[(long)(t * 16 + j + 8 * hi) * D + n] = acc[j] + bs;
}

// ---------------- launcher ----------------
extern "C" void kernel_launch(void* const* d_in, const int* in_sizes, int n_in,
                              void* d_out, int out_size, void* d_ws, size_t ws_size,
                              hipStream_t stream) {
    const float* x_source = (const float*)d_in[0];
    const float* x_target = (const float*)d_in[1];
    const int*   nb_rows  = (const int*)d_in[2];
    const int*   nb_cols  = (const int*)d_in[3];
    const float* nb_vals  = (const float*)d_in[4];
    const float* cci_vals = (const float*)d_in[5];
    const float* w_s      = (const float*)d_in[6];
    const float* w_t      = (const float*)d_in[7];
    const float* w_s_cci  = (const float*)d_in[8];
    const float* w_t_cci  = (const float*)d_in[9];
    const float* src_W    = (const float*)d_in[10];
    const float* src_b    = (const float*)d_in[11];
    const float* tgt_W    = (const float*)d_in[12];
    const float* tgt_b    = (const float*)d_in[13];
    float* ws  = (float*)d_ws;
    float* out = (float*)d_out;
    float* Bp  = ws + 16 * N64 + 2 * (long)N;

    long init_n = 6 * N64;
    k_init<<<dim3((unsigned)((init_n + 255) / 256)), 256, 0, stream>>>(ws);

    k_prep<<<dim3((2 * KTOT * D + 255) / 256), 256, 0, stream>>>(src_W, tgt_W, Bp);

    // 3125 row tiles, 4 waves/block, 4 matrices
    k_proj<<<dim3((N / 16 + 3) / 4, 4), 128, 0, stream>>>(
        x_source, x_target, w_s, w_t, w_s_cci, w_t_cci, ws);

    long sc_threads = (long)E * 32;
    k_scatter<<<dim3((unsigned)((sc_threads + 255) / 256)), 256, 0, stream>>>(
        nb_rows, nb_cols, nb_vals, cci_vals, ws);

    k_final<<<dim3(N / 16, 2), 128, 0, stream>>>(ws, src_b, tgt_b, out);
}